// Decoder_72335839200001
// MI455X (gfx1250) — compile-verified
//
#include <hip/hip_runtime.h>
#include <hip/hip_bf16.h>
#include <math.h>

#define N_NODES 20000
#define N_EDGES 400000
#define LATENT  128
#define H1      256      // HIDDEN/2
#define HIDDEN  512
#define OUT_D   64
#define HEADS   4
#define HD      (HEADS*OUT_D)   // 256
#define NEG_SLOPE 0.2f

typedef float v2f __attribute__((ext_vector_type(2)));
typedef float v8f __attribute__((ext_vector_type(8)));

// ---------------------------------------------------------------------------
// Weight repack: Bp[p][n] = (B[2p][n], B[2p+1][n]) stored as consecutive v2f.
// Lets the GEMM fetch each B fragment as one global_load_b64 from a single
// strided base pointer (weights are L2-resident, repack cost is negligible).
// ---------------------------------------------------------------------------
__global__ void repack_B(const float* __restrict__ B, float* __restrict__ Bp,
                         int K, int Nn) {
    int idx = blockIdx.x * blockDim.x + threadIdx.x;   // over (K/2)*Nn
    const int total = (K >> 1) * Nn;
    if (idx >= total) return;
    const int p = idx / Nn, n = idx % Nn;
    v2f v;
    v.x = B[(size_t)(2 * p) * Nn + n];
    v.y = B[(size_t)(2 * p + 1) * Nn + n];
    *(v2f*)(Bp + (size_t)p * 2 * Nn + 2 * n) = v;
}

// ---------------------------------------------------------------------------
// Tiled fp32 WMMA GEMM: C[M,N] = act(A[M,K] @ B[K,N] + bias), B pre-repacked.
// One wave computes a 16x64 tile via 4 accumulators, K-step of 4
// (V_WMMA_F32_16X16X4_F32). M % 16 == 0, N % 64 == 0, K % 4 == 0 here.
// A fragment: lanes 0-15 rows M=0..15, K pair {k0,k0+1}; lanes 16-31 K pair {k0+2,k0+3}.
// B fragment: lane holds col N with the matching K pair (one b64 from Bp).
// C fragment: VGPR r -> row r (lanes 0-15) / row r+8 (lanes 16-31), lane -> col.
// All control flow is wave-uniform, so EXEC is all-ones at every WMMA.
// ---------------------------------------------------------------------------
__global__ void gemm_wmma_f32(const float* __restrict__ A, const float* __restrict__ Bp,
                              const float* __restrict__ bias, float* __restrict__ C,
                              int M, int K, int Nn, int relu) {
    const int wavesPerBlock = blockDim.x >> 5;           // wave32
    const int wave   = blockIdx.x * wavesPerBlock + (threadIdx.x >> 5);
    const int lane   = threadIdx.x & 31;
    const int l      = lane & 15;
    const int half   = lane >> 4;                        // 0: K pair {k0,k0+1}, 1: {k0+2,k0+3}
    const int tilesM = M >> 4;
    const int tilesN = Nn >> 6;
    const int total  = tilesM * tilesN;
    const int stride = gridDim.x * wavesPerBlock;

    for (int t = wave; t < total; t += stride) {
        const int tm = t / tilesN, tn = t % tilesN;
        const int row0 = tm << 4, col0 = tn << 6;
        v8f acc0 = {}, acc1 = {}, acc2 = {}, acc3 = {};
        const float* Arow  = A + (size_t)(row0 + l) * K + (half << 1);
        // pair index for this lane group advances by k0/2; +half selects the pair
        const float* Bbase = Bp + (size_t)half * 2 * Nn + 2 * (col0 + l);

        for (int k0 = 0; k0 < K; k0 += 4) {
            v2f a = *(const v2f*)(Arow + k0);                     // global_load_b64
            const float* Bk = Bbase + (size_t)k0 * Nn;            // (k0/2)*2*Nn
            v2f b0 = *(const v2f*)(Bk + 0);                       // col0 + l
            v2f b1 = *(const v2f*)(Bk + 32);                      // +16 cols
            v2f b2 = *(const v2f*)(Bk + 64);                      // +32 cols
            v2f b3 = *(const v2f*)(Bk + 96);                      // +48 cols
            acc0 = __builtin_amdgcn_wmma_f32_16x16x4_f32(false, a, false, b0, (short)0, acc0, false, false);
            acc1 = __builtin_amdgcn_wmma_f32_16x16x4_f32(false, a, false, b1, (short)0, acc1, false, false);
            acc2 = __builtin_amdgcn_wmma_f32_16x16x4_f32(false, a, false, b2, (short)0, acc2, false, false);
            acc3 = __builtin_amdgcn_wmma_f32_16x16x4_f32(false, a, false, b3, (short)0, acc3, false, false);
        }

        v8f accs[4] = {acc0, acc1, acc2, acc3};
        #pragma unroll
        for (int j = 0; j < 4; ++j) {
            const int n = col0 + (j << 4) + l;
            const float bv = bias ? bias[n] : 0.f;
            #pragma unroll
            for (int r = 0; r < 8; ++r) {
                const int mrow = row0 + r + (half << 3);
                float v = accs[j][r] + bv;
                if (relu) v = fmaxf(v, 0.f);
                C[(size_t)mrow * Nn + n] = v;
            }
        }
    }
}

// a_src[n,h] = sum_d H[n,h,d]*att_src[h,d] ; same for a_dst
__global__ void att_coef(const float* __restrict__ Hm, const float* __restrict__ att_src,
                         const float* __restrict__ att_dst,
                         float* __restrict__ a_src, float* __restrict__ a_dst) {
    int idx = blockIdx.x * blockDim.x + threadIdx.x;
    if (idx >= N_NODES * HEADS) return;
    const int n = idx >> 2, h = idx & 3;
    const float* hp = Hm + (size_t)n * HD + h * OUT_D;
    const float* as = att_src + h * OUT_D;
    const float* ad = att_dst + h * OUT_D;
    float s = 0.f, d = 0.f;
    #pragma unroll 8
    for (int i = 0; i < OUT_D; ++i) { float v = hp[i]; s += v * as[i]; d += v * ad[i]; }
    a_src[idx] = s; a_dst[idx] = d;
}

__global__ void init_buffers(float* __restrict__ m, float* __restrict__ denom,
                             float* __restrict__ acc) {
    const int i = blockIdx.x * blockDim.x + threadIdx.x;
    const int stride = gridDim.x * blockDim.x;
    for (int j = i; j < N_NODES * HEADS; j += stride) { m[j] = -INFINITY; denom[j] = 0.f; }
    for (int j = i; j < N_NODES * HD;    j += stride) acc[j] = 0.f;
}

__device__ __forceinline__ void atomicMaxF(float* addr, float v) {
    if (v >= 0.f) atomicMax((int*)addr, __float_as_int(v));
    else          atomicMin((unsigned int*)addr, __float_as_uint(v));
}

__device__ __forceinline__ void edge_ends(const int* __restrict__ ei, int e, int& s, int& d) {
    if (e < N_EDGES) { s = ei[e]; d = ei[N_EDGES + e]; }
    else             { s = d = e - N_EDGES; }            // self loops
}

__global__ void edge_reduce_max(const int* __restrict__ ei, const float* __restrict__ a_src,
                                const float* __restrict__ a_dst, float* __restrict__ m) {
    int idx = blockIdx.x * blockDim.x + threadIdx.x;
    if (idx >= (N_EDGES + N_NODES) * HEADS) return;
    const int e = idx >> 2, h = idx & 3;
    int s, d; edge_ends(ei, e, s, d);
    float v = a_src[s * HEADS + h] + a_dst[d * HEADS + h];
    v = (v > 0.f) ? v : NEG_SLOPE * v;
    atomicMaxF(&m[d * HEADS + h], v);
}

__global__ void edge_exp_sum(const int* __restrict__ ei, const float* __restrict__ a_src,
                             const float* __restrict__ a_dst, const float* __restrict__ m,
                             float* __restrict__ e_exp, float* __restrict__ denom) {
    int idx = blockIdx.x * blockDim.x + threadIdx.x;
    if (idx >= (N_EDGES + N_NODES) * HEADS) return;
    const int e = idx >> 2, h = idx & 3;
    int s, d; edge_ends(ei, e, s, d);
    float v = a_src[s * HEADS + h] + a_dst[d * HEADS + h];
    v = (v > 0.f) ? v : NEG_SLOPE * v;
    const float ex = __expf(v - m[d * HEADS + h]);
    e_exp[idx] = ex;
    atomicAdd(&denom[d * HEADS + h], ex);
}

// One block (256 = HEADS*OUT_D threads) per edge: fully coalesced gather of H[src]
// and scatter-add into acc[dst] (atomics resolve in the 192MB L2).
__global__ void edge_message(const int* __restrict__ ei, const float* __restrict__ Hm,
                             const float* __restrict__ e_exp, const float* __restrict__ denom,
                             float* __restrict__ acc) {
    const int e = blockIdx.x;
    const int r = threadIdx.x;           // h = r/64, d = r%64
    const int h = r >> 6;
    int s, d; edge_ends(ei, e, s, d);
    const float w = e_exp[e * HEADS + h] / (denom[d * HEADS + h] + 1e-16f);
    const float v = Hm[(size_t)s * HD + r] * w;
    atomicAdd(&acc[(size_t)d * HD + r], v);
}

__global__ void finalize(const float* __restrict__ acc, const float* __restrict__ bias_g,
                         float* __restrict__ out) {
    int idx = blockIdx.x * blockDim.x + threadIdx.x;
    if (idx >= N_NODES * OUT_D) return;
    const int n = idx >> 6, d = idx & 63;
    const float* a = acc + (size_t)n * HD;
    out[idx] = 0.25f * (a[d] + a[OUT_D + d] + a[2 * OUT_D + d] + a[3 * OUT_D + d]) + bias_g[d];
}

extern "C" void kernel_launch(void* const* d_in, const int* in_sizes, int n_in,
                              void* d_out, int out_size, void* d_ws, size_t ws_size,
                              hipStream_t stream) {
    const float* z       = (const float*)d_in[0];
    const float* W1      = (const float*)d_in[1];
    const float* b1      = (const float*)d_in[2];
    const float* W2      = (const float*)d_in[3];
    const float* b2      = (const float*)d_in[4];
    const float* Wg      = (const float*)d_in[5];
    const float* att_src = (const float*)d_in[6];
    const float* att_dst = (const float*)d_in[7];
    const float* bias_g  = (const float*)d_in[8];
    const int*   ei      = (const int*)d_in[9];
    float* out = (float*)d_out;

    // Workspace layout (aliasing is safe: stream-ordered producers/consumers)
    char* ws = (char*)d_ws;
    size_t off = 0;
    auto carve = [&](size_t bytes) -> float* {
        float* p = (float*)(ws + off);
        off += (bytes + 255) & ~(size_t)255;
        return p;
    };
    float* x1    = carve((size_t)N_NODES * H1 * 4);      // dense1 out; later reused as e_exp
    float* x2    = carve((size_t)N_NODES * HIDDEN * 4);  // dense2 out; later reused as acc
    float* Hm    = carve((size_t)N_NODES * HD * 4);      // GAT projection
    float* W1p   = carve((size_t)LATENT * H1 * 4);       // repacked weights
    float* W2p   = carve((size_t)H1 * HIDDEN * 4);
    float* Wgp   = carve((size_t)HIDDEN * HD * 4);
    float* asrc  = carve((size_t)N_NODES * HEADS * 4);
    float* adst  = carve((size_t)N_NODES * HEADS * 4);
    float* mbuf  = carve((size_t)N_NODES * HEADS * 4);
    float* denom = carve((size_t)N_NODES * HEADS * 4);
    float* e_exp = x1;   // (E+N)*HEADS*4 = 6.7MB <= 20.5MB, x1 dead after GEMM2
    float* accb  = x2;   // N*HD*4 = 20.5MB <= 41MB,        x2 dead after GEMM3

    const int WPB = 8;   // waves per 256-thread block

    // Repack the three (small, L2-resident) weight matrices into K-pair v2f form.
    repack_B<<<((LATENT / 2) * H1     + 255) / 256, 256, 0, stream>>>(W1, W1p, LATENT, H1);
    repack_B<<<((H1 / 2) * HIDDEN    + 255) / 256, 256, 0, stream>>>(W2, W2p, H1, HIDDEN);
    repack_B<<<((HIDDEN / 2) * HD    + 255) / 256, 256, 0, stream>>>(Wg, Wgp, HIDDEN, HD);

    // dense1: relu(z @ W1 + b1)    [20000,128] x [128,256]
    {
        int waves = (N_NODES / 16) * (H1 / 64);
        gemm_wmma_f32<<<(waves + WPB - 1) / WPB, 256, 0, stream>>>(z, W1p, b1, x1,
                                                                   N_NODES, LATENT, H1, 1);
    }
    // dense2: relu(x1 @ W2 + b2)   [20000,256] x [256,512]
    {
        int waves = (N_NODES / 16) * (HIDDEN / 64);
        gemm_wmma_f32<<<(waves + WPB - 1) / WPB, 256, 0, stream>>>(x1, W2p, b2, x2,
                                                                   N_NODES, H1, HIDDEN, 1);
    }
    // GAT projection: x2 @ Wg      [20000,512] x [512,256]
    {
        int waves = (N_NODES / 16) * (HD / 64);
        gemm_wmma_f32<<<(waves + WPB - 1) / WPB, 256, 0, stream>>>(x2, Wgp, nullptr, Hm,
                                                                   N_NODES, HIDDEN, HD, 0);
    }

    att_coef<<<(N_NODES * HEADS + 255) / 256, 256, 0, stream>>>(Hm, att_src, att_dst, asrc, adst);
    init_buffers<<<4096, 256, 0, stream>>>(mbuf, denom, accb);

    const int T4 = (N_EDGES + N_NODES) * HEADS;
    edge_reduce_max<<<(T4 + 255) / 256, 256, 0, stream>>>(ei, asrc, adst, mbuf);
    edge_exp_sum<<<(T4 + 255) / 256, 256, 0, stream>>>(ei, asrc, adst, mbuf, e_exp, denom);
    edge_message<<<N_EDGES + N_NODES, HD, 0, stream>>>(ei, Hm, e_exp, denom, accb);
    finalize<<<(N_NODES * OUT_D + 255) / 256, 256, 0, stream>>>(accb, bias_g, out);
}